// SpectralCrossAttention_17952963297503
// MI455X (gfx1250) — compile-verified
//
#include <hip/hip_runtime.h>

typedef __attribute__((ext_vector_type(2))) float v2f;
typedef __attribute__((ext_vector_type(8))) float v8f;

#define NB   16          // batch
#define NC   256         // channels
#define HW   56          // spatial side
#define NP   3136        // 56*56
#define KDIM 256         // reduction dim (input channels)

// ---------------------------------------------------------------------------
// Channel-mixing GEMM via V_WMMA_F32_16X16X4_F32 (fp32 matrix cores).
// out[b][m][p] = sum_c W[m][c] * in[b][c][p]  (+ bias[m])
// Each wave: one 16(M) x 64(P) strip = 4 independent 16x16 accumulator
// chains, K=256 in steps of 4 -> 4x64 WMMA per wave.  A-fragment comes from
// a 16x256 LDS weight tile shared by the 8 waves of the block; B-fragments
// come straight from global via one pointer + immediate offsets.
// ---------------------------------------------------------------------------
__global__ __launch_bounds__(256)
void gemm_cmix_wmma(const float* __restrict__ W,     // [M][256] row-major
                    const float* __restrict__ in,    // [B][256][P]
                    float*       __restrict__ out,   // [B][M][P]
                    const float* __restrict__ bias,  // [M] or nullptr
                    int M)
{
    __shared__ float Wt[16 * KDIM];

    const int b  = blockIdx.z;
    const int m0 = blockIdx.y * 16;

    // cooperative load of the 16 x 256 weight tile
    for (int t = threadIdx.x; t < 16 * KDIM; t += 256)
        Wt[t] = W[(size_t)(m0 + (t >> 8)) * KDIM + (t & 255)];
    __syncthreads();

    const int wave = threadIdx.x >> 5;
    const int lane = threadIdx.x & 31;
    const int grp  = blockIdx.x * 8 + wave;   // group of 4 N-tiles (64 cols)
    const int nGroups = NP / 64;              // 49
    if (grp >= nGroups) return;               // whole-wave uniform exit

    const int p0   = grp * 64;
    const int half = lane >> 4;               // 0 = lanes 0-15, 1 = lanes 16-31
    const int lm   = lane & 15;

    // B pointer: row (2*half), column p0+lm; rows advance by NP
    const float* bp = in + (size_t)b * KDIM * NP + (size_t)(2 * half) * NP + p0 + lm;
    const float* wp = &Wt[lm * KDIM + 2 * half];
    float*       outb = out + ((size_t)b * M + m0) * NP + p0;

    v8f acc0 = {}, acc1 = {}, acc2 = {}, acc3 = {};

    for (int k0 = 0; k0 < KDIM; k0 += 4) {
        // near-cache prefetch 8 K-rows ahead (address clamped, branch-free)
        const size_t pfo = (k0 < KDIM - 8) ? (size_t)(8 * NP) : 0;
        __builtin_prefetch(bp + pfo, 0, 3);

        // A fragment (16x4): lane=M, VGPR0:K0/K2, VGPR1:K1/K3  (ds_load_b64)
        v2f a;
        a.x = wp[0];
        a.y = wp[1];

        // 4 B fragments (4x16 each): one base, immediate offsets
        v2f b0, b1, b2, b3;
        b0.x = bp[0];       b0.y = bp[NP];
        b1.x = bp[16];      b1.y = bp[NP + 16];
        b2.x = bp[32];      b2.y = bp[NP + 32];
        b3.x = bp[48];      b3.y = bp[NP + 48];

        acc0 = __builtin_amdgcn_wmma_f32_16x16x4_f32(false, a, false, b0,
                                                     (short)0, acc0, false, false);
        acc1 = __builtin_amdgcn_wmma_f32_16x16x4_f32(false, a, false, b1,
                                                     (short)0, acc1, false, false);
        acc2 = __builtin_amdgcn_wmma_f32_16x16x4_f32(false, a, false, b2,
                                                     (short)0, acc2, false, false);
        acc3 = __builtin_amdgcn_wmma_f32_16x16x4_f32(false, a, false, b3,
                                                     (short)0, acc3, false, false);

        wp += 4;
        bp += (size_t)4 * NP;
    }

    // C/D layout: VGPR i -> M = i (lanes 0-15) / i+8 (lanes 16-31), N = lane%16
#pragma unroll
    for (int i = 0; i < 8; ++i) {
        const int row = i + 8 * half;
        const float bv = bias ? bias[m0 + row] : 0.0f;
        float* orow = outb + (size_t)row * NP + lm;
        orow[0]  = acc0[i] + bv;
        orow[16] = acc1[i] + bv;
        orow[32] = acc2[i] + bv;
        orow[48] = acc3[i] + bv;
    }
}

// ---------------------------------------------------------------------------
// Fused spectral attention + depthwise conv, one workgroup per (b,c) plane.
// attn = irfft2(rfft2(q)*rfft2(k), ortho) = circconv2(q,k)/56
// out_mid = attn * v + dwconv3x3(v) + dw_b          (written in-place over q)
// All DFT passes run out of LDS (plane = 12.5KB, complex = 25KB).
// ---------------------------------------------------------------------------
__global__ __launch_bounds__(256)
void spectral_fuse(float*       __restrict__ qmid,   // [B*C][P]  q in / mid out
                   const float* __restrict__ kvbuf,  // [B][512][P]
                   const float* __restrict__ dw_w,   // [C][9]
                   const float* __restrict__ dw_b)   // [C]
{
    __shared__ float2 tw[NP];    // e^{-2*pi*i*j*k/56}, indexed [j*56+k]
    __shared__ float  R[NP];     // real staging (q, then k, then v)
    __shared__ float2 Af[NP];
    __shared__ float2 Bf[NP];
    __shared__ float  dwv[10];   // 9 taps + bias

    const int pi  = blockIdx.x;          // b*256 + c
    const int b   = pi >> 8;
    const int c   = pi & 255;
    const int tid = threadIdx.x;

    float*       qplane = qmid + (size_t)pi * NP;
    const float* kplane = kvbuf + ((size_t)b * 512 + c)       * NP;
    const float* vplane = kvbuf + ((size_t)b * 512 + 256 + c) * NP;

    // S1: twiddle table + load q plane
    for (int i = tid; i < NP; i += 256) {
        const int j = i / HW, kk = i % HW;
        const int t = (j * kk) % HW;             // exact periodic reduction
        float s, co;
        __sincosf(-6.283185307179586f * (float)t / (float)HW, &s, &co);
        float2 w; w.x = co; w.y = s;
        tw[i] = w;
        R[i]  = qplane[i];
    }
    __syncthreads();

    // S2: q row DFT:  Af[r][f] = sum_c R[r][c] * tw[f][c]
    for (int i = tid; i < NP; i += 256) {
        const int r = i / HW, f = i % HW;
        float re = 0.f, im = 0.f;
        const float*  rp = &R[r * HW];
        const float2* wp = &tw[f * HW];
#pragma unroll 8
        for (int cc = 0; cc < HW; ++cc) {
            const float v = rp[cc]; const float2 w = wp[cc];
            re = fmaf(v, w.x, re);
            im = fmaf(v, w.y, im);
        }
        float2 o; o.x = re; o.y = im; Af[i] = o;
    }
    __syncthreads();

    // S3: q col DFT into Bf; overlap loading k plane into R
    for (int i = tid; i < NP; i += 256) {
        const int fr = i / HW, fc = i % HW;
        float re = 0.f, im = 0.f;
        const float2* wp = &tw[fr * HW];
#pragma unroll 8
        for (int r = 0; r < HW; ++r) {
            const float2 z = Af[r * HW + fc]; const float2 w = wp[r];
            re += z.x * w.x - z.y * w.y;
            im += z.x * w.y + z.y * w.x;
        }
        float2 o; o.x = re; o.y = im; Bf[i] = o;
        R[i] = kplane[i];
    }
    __syncthreads();

    // S4: k row DFT -> Af
    for (int i = tid; i < NP; i += 256) {
        const int r = i / HW, f = i % HW;
        float re = 0.f, im = 0.f;
        const float*  rp = &R[r * HW];
        const float2* wp = &tw[f * HW];
#pragma unroll 8
        for (int cc = 0; cc < HW; ++cc) {
            const float v = rp[cc]; const float2 w = wp[cc];
            re = fmaf(v, w.x, re);
            im = fmaf(v, w.y, im);
        }
        float2 o; o.x = re; o.y = im; Af[i] = o;
    }
    __syncthreads();

    // S5: k col DFT, pointwise product into Bf; overlap loading v + dw taps
    if (tid < 9)  dwv[tid] = dw_w[(size_t)c * 9 + tid];
    if (tid == 9) dwv[9]   = dw_b[c];
    for (int i = tid; i < NP; i += 256) {
        const int fr = i / HW, fc = i % HW;
        float re = 0.f, im = 0.f;
        const float2* wp = &tw[fr * HW];
#pragma unroll 8
        for (int r = 0; r < HW; ++r) {
            const float2 z = Af[r * HW + fc]; const float2 w = wp[r];
            re += z.x * w.x - z.y * w.y;
            im += z.x * w.y + z.y * w.x;
        }
        const float2 qf = Bf[i];
        float2 o;
        o.x = qf.x * re - qf.y * im;
        o.y = qf.x * im + qf.y * re;
        Bf[i] = o;
        R[i] = vplane[i];
    }
    __syncthreads();

    // S6: inverse row DFT (conj twiddles): Af[fr][c] = sum_fc Bf[fr][fc]*conj(tw[c][fc])
    for (int i = tid; i < NP; i += 256) {
        const int fr = i / HW, cc = i % HW;
        float re = 0.f, im = 0.f;
        const float2* wp = &tw[cc * HW];
#pragma unroll 8
        for (int fc = 0; fc < HW; ++fc) {
            const float2 z = Bf[fr * HW + fc]; const float2 w = wp[fc];
            re += z.x * w.x + z.y * w.y;
            im += z.y * w.x - z.x * w.y;
        }
        float2 o; o.x = re; o.y = im; Af[i] = o;
    }
    __syncthreads();

    // S7: inverse col DFT (real part) + fuse: attn*v + dwconv(v) + bias
    const float scale = 1.0f / 175616.0f;   // 1/(56*56) IDFT * 1/56 ortho
    for (int i = tid; i < NP; i += 256) {
        const int r = i / HW, cc = i % HW;
        float re = 0.f;
        const float2* wp = &tw[r * HW];
#pragma unroll 8
        for (int fr = 0; fr < HW; ++fr) {
            const float2 z = Af[fr * HW + cc]; const float2 w = wp[fr];
            re += z.x * w.x + z.y * w.y;      // Re(z * conj(w))
        }
        const float attn = re * scale;

        float conv = dwv[9];
#pragma unroll
        for (int dy = -1; dy <= 1; ++dy)
#pragma unroll
            for (int dx = -1; dx <= 1; ++dx) {
                const int rr = r + dy, c2 = cc + dx;
                if (rr >= 0 && rr < HW && c2 >= 0 && c2 < HW)
                    conv = fmaf(R[rr * HW + c2], dwv[(dy + 1) * 3 + (dx + 1)], conv);
            }

        qplane[i] = attn * R[i] + conv;
    }
}

// ---------------------------------------------------------------------------
extern "C" void kernel_launch(void* const* d_in, const int* in_sizes, int n_in,
                              void* d_out, int out_size, void* d_ws, size_t ws_size,
                              hipStream_t stream)
{
    const float* x       = (const float*)d_in[0];
    const float* context = (const float*)d_in[1];
    const float* Wq      = (const float*)d_in[2];
    const float* Wkv     = (const float*)d_in[3];
    const float* Wproj   = (const float*)d_in[4];
    const float* bproj   = (const float*)d_in[5];
    const float* dw_w    = (const float*)d_in[6];
    const float* dw_b    = (const float*)d_in[7];
    float*       out     = (float*)d_out;

    // workspace: q/mid buffer [B][256][P], kv buffer [B][512][P]  (~154 MB)
    float* qbuf  = (float*)d_ws;
    float* kvbuf = qbuf + (size_t)NB * NC * NP;

    const dim3 blk(256);
    const dim3 gq (7, 16, NB);    // 49 N-groups / 8 waves, 256/16 M-tiles
    const dim3 gkv(7, 32, NB);    // 512/16 M-tiles

    gemm_cmix_wmma<<<gq,  blk, 0, stream>>>(Wq,  x,       qbuf,  nullptr, 256);
    gemm_cmix_wmma<<<gkv, blk, 0, stream>>>(Wkv, context, kvbuf, nullptr, 512);
    spectral_fuse <<<dim3(NB * NC), blk, 0, stream>>>(qbuf, kvbuf, dw_w, dw_b);
    gemm_cmix_wmma<<<gq,  blk, 0, stream>>>(Wproj, qbuf,  out,   bproj,  256);
}